// OuterProductLinear_80341658239350
// MI455X (gfx1250) — compile-verified
//
#include <hip/hip_runtime.h>

typedef __attribute__((ext_vector_type(2))) float v2f;
typedef __attribute__((ext_vector_type(8))) float v8f;

#define DIM    1024
#define BATCH  256
#define NCHUNK 8    // column chunks (blockIdx.y)
#define WAVES  8    // wave32 per block

// Main fused kernel:
//   T1 = cf @ W_mat, T2 = |cf| @ W_mat   (two f32 WMMA accumulators, shared B)
//   s[b, j] = 0.55*T1[b,j]*pf[b,j] + 0.45*T2[b,j]*|pf[b,j]|
//   partial[chunk, b] = Σ_{j in chunk} s[b, j]
__global__ __launch_bounds__(256)
void OuterProductLinear_main(const float* __restrict__ cf,
                             const float* __restrict__ pf,
                             const float* __restrict__ W,
                             float* __restrict__ partial) {
    __shared__ float red[WAVES][16];

    const int tid  = threadIdx.x;
    const int wv   = tid >> 5;          // wave id 0..7
    const int lane = tid & 31;
    const int col  = lane & 15;         // M for A-matrix, N for B-matrix
    const int kk   = (lane >> 4) << 1;  // K sub-offset: 0 (lanes 0-15) / 2 (lanes 16-31)

    const int m0 = blockIdx.x * 16;                          // batch tile
    const int n0 = (blockIdx.y * WAVES + wv) * 16;           // column tile

    // A: 16x4 f32 tile of cf   -> lane holds (K=kk, K=kk+1) of row m0+col
    // B: 4x16 f32 tile of W    -> lane holds (K=kk, K=kk+1) of col n0+col
    const float* aPtr = cf + (m0 + col) * DIM + kk;
    const float* bPtr = W  + kk * DIM + n0 + col;

    v8f c1 = {};   // cf  @ W
    v8f c2 = {};   // |cf| @ W
    #pragma unroll 4
    for (int k = 0; k < DIM; k += 4) {
        v2f a = *(const v2f*)aPtr;               // cf[m0+col, k+kk .. k+kk+1]
        v2f b;
        b.x = bPtr[0];                           // W[k+kk,   n0+col]
        b.y = bPtr[DIM];                         // W[k+kk+1, n0+col]
        v2f aa;
        aa.x = __builtin_fabsf(a.x);
        aa.y = __builtin_fabsf(a.y);
        // D = A*B + C  (v_wmma_f32_16x16x4_f32)
        c1 = __builtin_amdgcn_wmma_f32_16x16x4_f32(false, a,  false, b,
                                                   (short)0, c1, false, false);
        c2 = __builtin_amdgcn_wmma_f32_16x16x4_f32(false, aa, false, b,
                                                   (short)0, c2, false, false);
        aPtr += 4;
        bPtr += 4 * DIM;
    }

    // C/D layout: lane holds rows M = r + mrow (mrow = 0 for lanes 0-15, 8 for 16-31),
    // column N = lane&15, for VGPR r = 0..7.
    const int mrow = (lane >> 4) << 3;
    float acc[8];
    #pragma unroll
    for (int r = 0; r < 8; ++r) {
        float p = pf[(m0 + mrow + r) * DIM + n0 + col];
        acc[r] = 0.55f * c1[r] * p + 0.45f * c2[r] * __builtin_fabsf(p);
    }

    // Reduce over the 16 columns of the tile: xor masks 1..8 stay inside each
    // 16-lane half of the wave32, so the two row groups reduce independently.
    #pragma unroll
    for (int r = 0; r < 8; ++r) {
        float v = acc[r];
        v += __shfl_xor(v, 1);
        v += __shfl_xor(v, 2);
        v += __shfl_xor(v, 4);
        v += __shfl_xor(v, 8);
        acc[r] = v;
    }

    if (col == 0) {  // lane 0 -> rows 0..7, lane 16 -> rows 8..15
        #pragma unroll
        for (int r = 0; r < 8; ++r) red[wv][mrow + r] = acc[r];
    }
    __syncthreads();

    // Cross-wave reduction: 8 waves cover 8 distinct column tiles of this chunk.
    if (tid < 16) {
        float s = 0.0f;
        #pragma unroll
        for (int w = 0; w < WAVES; ++w) s += red[w][tid];
        partial[blockIdx.y * BATCH + m0 + tid] = s;
    }
}

// Fold the NCHUNK deterministic partials + bias into the final (256,1) output.
__global__ __launch_bounds__(256)
void OuterProductLinear_finalize(const float* __restrict__ partial,
                                 const float* __restrict__ bias,
                                 float* __restrict__ out) {
    int b = blockIdx.x * blockDim.x + threadIdx.x;
    if (b < BATCH) {
        float s = bias[0];
        #pragma unroll
        for (int c = 0; c < NCHUNK; ++c) s += partial[c * BATCH + b];
        out[b] = s;
    }
}

extern "C" void kernel_launch(void* const* d_in, const int* in_sizes, int n_in,
                              void* d_out, int out_size, void* d_ws, size_t ws_size,
                              hipStream_t stream) {
    const float* cf   = (const float*)d_in[0];   // (256, 1024) f32
    const float* pf   = (const float*)d_in[1];   // (256, 1024) f32
    const float* W    = (const float*)d_in[2];   // (1, 1024*1024) f32
    const float* bias = (const float*)d_in[3];   // (1,) f32
    float* out = (float*)d_out;                  // (256, 1) f32

    float* partial = (float*)d_ws;               // NCHUNK * 256 floats = 8 KB

    dim3 grid(BATCH / 16, NCHUNK);               // 16 batch tiles x 8 column chunks
    OuterProductLinear_main<<<grid, 256, 0, stream>>>(cf, pf, W, partial);
    OuterProductLinear_finalize<<<1, 256, 0, stream>>>(partial, bias, out);
}